// MDCLBR_55774445306557
// MI455X (gfx1250) — compile-verified
//
#include <hip/hip_runtime.h>
#include <hip/hip_bf16.h>

// ---------------------------------------------------------------------------
// BGCN-style propagation for MI455X (gfx1250, wave32).
//  - SpMM: edge-parallel COO scatter, 16 lanes/edge, float4 gathers, native
//    fp32 global atomics (resolve in 192MB L2). Edge metadata staged into LDS
//    via gfx1250 async global->LDS copies (ASYNCcnt) when available.
//  - Row L2 norms + acc update: one wave per 16 rows, squared norms computed
//    on the matrix pipe as diag(X*X^T) via v_wmma_f32_16x16x4_f32.
// ---------------------------------------------------------------------------

#define NU 50000
#define NI 40000
#define NB 20000
#define ND 64

typedef __attribute__((ext_vector_type(2))) float v2f;
typedef __attribute__((ext_vector_type(8))) float v8f;

#if __has_builtin(__builtin_amdgcn_global_load_async_to_lds_b32) && \
    __has_builtin(__builtin_amdgcn_s_wait_asynccnt)
#define HAVE_ASYNC_LDS 1
__device__ __forceinline__ void async_copy_b32(const void* gsrc, void* ldst) {
    __builtin_amdgcn_global_load_async_to_lds_b32(
        (__attribute__((address_space(1))) int*)(gsrc),
        (__attribute__((address_space(3))) int*)(ldst),
        /*imm offset=*/0, /*cpol=*/0);
}
#else
#define HAVE_ASYNC_LDS 0
#endif

// feats/acc = concat(a, b) ; operates on float4 granularity
__global__ void concat_init_k(const float4* __restrict__ a,
                              const float4* __restrict__ b,
                              float4* __restrict__ feats,
                              float4* __restrict__ acc,
                              int na4, int ntot4) {
    int i = blockIdx.x * blockDim.x + threadIdx.x;
    if (i < ntot4) {
        float4 v = (i < na4) ? a[i] : b[i - na4];
        feats[i] = v;
        acc[i]   = v;
    }
}

// y[rows[e]] += scale * vals[e] * x[cols[e]]  (16 lanes per edge, float4 each)
// Block owns 16 edges; edge metadata streamed to LDS via async DMA.
__global__ __launch_bounds__(256) void spmm_atomic_k(const int* __restrict__ rows,
                                                     const int* __restrict__ cols,
                                                     const float* __restrict__ vals,
                                                     const float* __restrict__ x,
                                                     float* __restrict__ y,
                                                     int nnz, float scale) {
    const int t  = threadIdx.x;
    const int e0 = blockIdx.x * 16;
    const int el = t >> 4;   // local edge 0..15
    const int l  = t & 15;   // float4 slot within the 64-float row
    const int e  = e0 + el;

    int   r, c;
    float v;
#if HAVE_ASYNC_LDS
    __shared__ int   s_rows[16];
    __shared__ int   s_cols[16];
    __shared__ float s_vals[16];
    if (t < 16 && (e0 + t) < nnz) {
        async_copy_b32(&rows[e0 + t], &s_rows[t]);
        async_copy_b32(&cols[e0 + t], &s_cols[t]);
        async_copy_b32(&vals[e0 + t], &s_vals[t]);
    }
    __builtin_amdgcn_s_wait_asynccnt(0);
    __syncthreads();
    if (e >= nnz) return;
    r = s_rows[el];
    c = s_cols[el];
    v = s_vals[el] * scale;
#else
    if (e >= nnz) return;
    r = rows[e];
    c = cols[e];
    v = vals[e] * scale;
#endif
    const float4 xv = *(const float4*)(x + (size_t)c * ND + l * 4);
    float* yp = y + (size_t)r * ND + l * 4;
    unsafeAtomicAdd(yp + 0, v * xv.x);
    unsafeAtomicAdd(yp + 1, v * xv.y);
    unsafeAtomicAdd(yp + 2, v * xv.z);
    unsafeAtomicAdd(yp + 3, v * xv.w);
}

// For 16 rows: sq-norms = diag(X * X^T) via 16x v_wmma_f32_16x16x4_f32,
// then acc[row] += x[row] / max(||x[row]||, 1e-12).
// grid.x = n/16 (n must be a multiple of 16), block = 32 (one wave).
// A-matrix 16x4 f32 layout: lanes 0-15 hold M=0..15 with VGPR0=K0,VGPR1=K1;
// lanes 16-31 hold K2,K3.  Feeding the same registers as B (4x16 = (X chunk)^T)
// yields D = X * X^T accumulated over K.
__global__ __launch_bounds__(32) void norm_acc_wmma_k(const float* __restrict__ xin,
                                                      float* __restrict__ acc) {
    const int lane = threadIdx.x;          // 0..31
    const int m    = lane & 15;            // row within 16-row tile
    const int half = lane >> 4;            // 0: K0/K1, 1: K2/K3
    const size_t row = (size_t)blockIdx.x * 16 + m;
    const float* xr = xin + row * ND;
    float*       ar = acc + row * ND;

    v8f c = {0.f, 0.f, 0.f, 0.f, 0.f, 0.f, 0.f, 0.f};
    v2f xv[16];
#pragma unroll
    for (int kk = 0; kk < 16; ++kk) {
        const int o = kk * 4 + half * 2;
        v2f a;
        a.x = xr[o];
        a.y = xr[o + 1];
        xv[kk] = a;
        // D = A*B + C ; A == B registers -> Gram matrix accumulation
        c = __builtin_amdgcn_wmma_f32_16x16x4_f32(
                /*neg_a=*/false, a, /*neg_b=*/false, a,
                /*c_mod=*/(short)0, c, /*reuse_a=*/false, /*reuse_b=*/false);
    }

    // C/D layout: VGPR r -> lanes 0-15: (M=r, N=lane); lanes 16-31: (M=r+8, N=lane-16)
    // diag(M=N): row r sits in c[r] at lane r (r<8) / c[r-8] at lane r+16 (r>=8).
    // Each *source* lane pre-selects the diagonal value it owns, then one
    // bpermute routes it to every consumer lane.
    const int idx = lane & 7;                 // which c[] this lane supplies
    float supply = c[0];
#pragma unroll
    for (int rr = 1; rr < 8; ++rr) supply = (idx == rr) ? c[rr] : supply;
    const int src = (m < 8) ? m : (m + 16);   // lane holding diag of row m
    const float sq = __shfl(supply, src, 32);

    const float inv = 1.0f / fmaxf(sqrtf(sq), 1e-12f);
#pragma unroll
    for (int kk = 0; kk < 16; ++kk) {
        const int o = kk * 4 + half * 2;
        ar[o]     += xv[kk].x * inv;
        ar[o + 1] += xv[kk].y * inv;
    }
}

static inline int cdiv(long a, long b) { return (int)((a + b - 1) / b); }

extern "C" void kernel_launch(void* const* d_in, const int* in_sizes, int n_in,
                              void* d_out, int out_size, void* d_ws, size_t ws_size,
                              hipStream_t stream) {
    const float* users   = (const float*)d_in[0];
    const float* items   = (const float*)d_in[1];
    const float* bundles = (const float*)d_in[2];
    const int*   il_rows = (const int*)d_in[3];
    const int*   il_cols = (const int*)d_in[4];
    const float* il_vals = (const float*)d_in[5];
    const int    il_nnz  = in_sizes[3];
    const int*   bl_rows = (const int*)d_in[6];
    const int*   bl_cols = (const int*)d_in[7];
    const float* bl_vals = (const float*)d_in[8];
    const int    bl_nnz  = in_sizes[6];
    const int*   bi_rows = (const int*)d_in[9];
    const int*   bi_cols = (const int*)d_in[10];
    const float* bi_vals = (const float*)d_in[11];
    const int    bi_nnz  = in_sizes[9];

    float* out = (float*)d_out;
    const size_t N_IL = (size_t)(NU + NI);   // 90000, multiple of 16
    const size_t N_BL = (size_t)(NU + NB);   // 70000, multiple of 16

    // workspace: A | B | acc, each N_IL*64 floats (reused for bundle level)
    float* A   = (float*)d_ws;
    float* Bf  = A  + N_IL * ND;
    float* acc = Bf + N_IL * ND;

    // d_out layout: il_users[U] | bl_users[U] | il_bundles[B] | bl_bundles[B]
    float* il_users_out   = out;
    float* bl_users_out   = out + (size_t)NU * ND;
    float* il_bundles_out = out + (size_t)2 * NU * ND;
    float* bl_bundles_out = out + (size_t)2 * NU * ND + (size_t)NB * ND;

    const int TB = 256;

    // ---------------- item-level propagation (n = U + I) ----------------
    {
        const int ntot4 = (int)(N_IL * ND / 4);
        const int na4   = NU * ND / 4;
        concat_init_k<<<cdiv(ntot4, TB), TB, 0, stream>>>(
            (const float4*)users, (const float4*)items,
            (float4*)A, (float4*)acc, na4, ntot4);

        // layer 0: B = SpMM(A) / 2 ; acc += B / ||B||
        (void)hipMemsetAsync(Bf, 0, N_IL * ND * sizeof(float), stream);
        spmm_atomic_k<<<cdiv(il_nnz, 16), TB, 0, stream>>>(
            il_rows, il_cols, il_vals, A, Bf, il_nnz, 0.5f);
        norm_acc_wmma_k<<<(int)(N_IL / 16), 32, 0, stream>>>(Bf, acc);

        // layer 1: A = SpMM(B) / 3 ; acc += A / ||A||
        (void)hipMemsetAsync(A, 0, N_IL * ND * sizeof(float), stream);
        spmm_atomic_k<<<cdiv(il_nnz, 16), TB, 0, stream>>>(
            il_rows, il_cols, il_vals, Bf, A, il_nnz, 1.0f / 3.0f);
        norm_acc_wmma_k<<<(int)(N_IL / 16), 32, 0, stream>>>(A, acc);
    }

    // bundle aggregation: il_bundles = BI @ acc_items (row-normalized weights)
    (void)hipMemsetAsync(il_bundles_out, 0, (size_t)NB * ND * sizeof(float), stream);
    spmm_atomic_k<<<cdiv(bi_nnz, 16), TB, 0, stream>>>(
        bi_rows, bi_cols, bi_vals, acc + (size_t)NU * ND, il_bundles_out,
        bi_nnz, 1.0f);

    // il_users -> out
    (void)hipMemcpyAsync(il_users_out, acc, (size_t)NU * ND * sizeof(float),
                         hipMemcpyDeviceToDevice, stream);

    // ---------------- bundle-level propagation (n = U + B) ----------------
    {
        const int ntot4 = (int)(N_BL * ND / 4);
        const int na4   = NU * ND / 4;
        concat_init_k<<<cdiv(ntot4, TB), TB, 0, stream>>>(
            (const float4*)users, (const float4*)bundles,
            (float4*)A, (float4*)acc, na4, ntot4);

        (void)hipMemsetAsync(Bf, 0, N_BL * ND * sizeof(float), stream);
        spmm_atomic_k<<<cdiv(bl_nnz, 16), TB, 0, stream>>>(
            bl_rows, bl_cols, bl_vals, A, Bf, bl_nnz, 0.5f);
        norm_acc_wmma_k<<<(int)(N_BL / 16), 32, 0, stream>>>(Bf, acc);

        (void)hipMemsetAsync(A, 0, N_BL * ND * sizeof(float), stream);
        spmm_atomic_k<<<cdiv(bl_nnz, 16), TB, 0, stream>>>(
            bl_rows, bl_cols, bl_vals, Bf, A, bl_nnz, 1.0f / 3.0f);
        norm_acc_wmma_k<<<(int)(N_BL / 16), 32, 0, stream>>>(A, acc);
    }

    // bl_users, bl_bundles -> out
    (void)hipMemcpyAsync(bl_users_out, acc, (size_t)NU * ND * sizeof(float),
                         hipMemcpyDeviceToDevice, stream);
    (void)hipMemcpyAsync(bl_bundles_out, acc + (size_t)NU * ND,
                         (size_t)NB * ND * sizeof(float),
                         hipMemcpyDeviceToDevice, stream);
}